// customLoss_34110630265423
// MI455X (gfx1250) — compile-verified
//
#include <hip/hip_runtime.h>
#include <hip/hip_bf16.h>

// ---------------------------------------------------------------------------
// Custom loss: P*L1 + (1-P)*CE over [B=2048, C=50257] fp32 logits.
// Memory-bound: 412 MB read once -> ~17.7 us floor at 23.3 TB/s.
// Kernel 1: one block per row, single streaming pass (online softmax + sums).
// Kernel 2: one wave, WMMA (v_wmma_f32_16x16x4_f32) all-ones accumulator
//           reduces the 2048 per-row losses to the scalar output.
// ---------------------------------------------------------------------------

typedef float v2f __attribute__((ext_vector_type(2)));
typedef float v4f __attribute__((ext_vector_type(4)));
typedef float v8f __attribute__((ext_vector_type(8)));

namespace {
constexpr int   kB       = 2048;
constexpr int   kC       = 50257;
constexpr int   kBlock   = 256;          // 8 waves (wave32)
constexpr int   kWaves   = kBlock / 32;
constexpr float kScaleT  = 10.0f;        // SCALE_TRUE
// SCALE_WRONG = -10 is folded into the closed forms below.
}

// Per-element L1 contribution for a NON-label column:
//   tempOut = x>0 ? -10x : 0 ; tempLabel = -10
//   |tempOut - tempLabel| = x>0 ? 10*|1-x| : 10
__device__ __forceinline__ float l1_elem(float x) {
    return (x > 0.0f) ? kScaleT * fabsf(1.0f - x) : kScaleT;
}

__global__ __launch_bounds__(kBlock)
void rowloss_kernel(const float* __restrict__ logits,
                    const int*  __restrict__ labels,
                    float*      __restrict__ row_loss) {
    const int row = blockIdx.x;
    const int tid = threadIdx.x;
    const float* __restrict__ rowp = logits + (size_t)row * kC;
    const int label = labels[row];

    __shared__ float s_m[kWaves], s_s[kWaves], s_sx[kWaves], s_l1[kWaves];
    __shared__ float s_xl;

    // Per-lane online-softmax state + sums.
    float m  = -3.4028235e38f;   // running max
    float s  = 0.0f;             // running sum exp(x - m)
    float sx = 0.0f;             // running sum x
    float l1 = 0.0f;             // running L1 partial (generic form everywhere)

    const int nvec = kC >> 2;    // 12564 float4's; element kC-1 is the tail
    const v4f* __restrict__ rowv = (const v4f*)rowp;

    for (int i = tid; i < nvec; i += kBlock) {
        v4f v = __builtin_nontemporal_load(rowv + i);   // read-once -> NT hint
        unsigned d = (unsigned)(label - (i << 2));
        if (d < 4u) s_xl = v[d];                        // capture gathered logit
#pragma unroll
        for (int j = 0; j < 4; ++j) {
            float x = v[j];
            sx += x;
            l1 += l1_elem(x);
            if (x <= m) {                               // common path: 1 exp
                s += __expf(x - m);
            } else {                                    // rare: max update
                s = s * __expf(m - x) + 1.0f;
                m = x;
            }
        }
    }

    // Tail element (index kC-1), handled by thread 0 only.
    if (tid == 0) {
        float x = rowp[kC - 1];
        if (label == kC - 1) s_xl = x;
        sx += x;
        l1 += l1_elem(x);
        if (x <= m) s += __expf(x - m);
        else { s = s * __expf(m - x) + 1.0f; m = x; }
    }

    // Wave32 reduction (fixed order -> deterministic).
#pragma unroll
    for (int off = 16; off > 0; off >>= 1) {
        float m2 = __shfl_xor(m, off, 32);
        float s2 = __shfl_xor(s, off, 32);
        float mn = fmaxf(m, m2);
        s  = s * __expf(m - mn) + s2 * __expf(m2 - mn);
        m  = mn;
        sx += __shfl_xor(sx, off, 32);
        l1 += __shfl_xor(l1, off, 32);
    }
    const int wave = tid >> 5;
    if ((tid & 31) == 0) {
        s_m[wave] = m; s_s[wave] = s; s_sx[wave] = sx; s_l1[wave] = l1;
    }
    __syncthreads();

    if (tid == 0) {
        float M = s_m[0], S = s_s[0], SX = s_sx[0], L1 = s_l1[0];
#pragma unroll
        for (int w = 1; w < kWaves; ++w) {
            float mw = s_m[w], sw = s_s[w];
            float mn = fmaxf(M, mw);
            S = S * __expf(M - mn) + sw * __expf(mw - mn);
            M = mn;
            SX += s_sx[w];
            L1 += s_l1[w];
        }
        const float xl = s_xl;
        // Label-column fixup for L1:
        //   tempOut(xl) = xl>0 ? -10*xl : 0   (<= 0)
        //   row_val     = 10*max(|mean|,|xl|) (>= 0)
        //   |tempOut - row_val| = row_val + (xl>0 ? 10*xl : 0)
        const float mean    = SX * (1.0f / (float)kC);
        const float row_val = kScaleT * fmaxf(fabsf(mean), fabsf(xl));
        const float g       = row_val + ((xl > 0.0f) ? kScaleT * xl : 0.0f);
        const float l1_row  = L1 - l1_elem(xl) + g;
        // CE: -(xl - M - log(S))
        const float ce_row  = M + __logf(S) - xl;
        row_loss[row] = 0.5f * l1_row * (1.0f / ((float)kB * (float)kC))
                      + 0.5f * ce_row * (1.0f / (float)kB);
    }
}

// ---------------------------------------------------------------------------
// Final reduction: single wave32, V_WMMA_F32_16X16X4_F32 with A == ones(16x4):
//   D[m][n] = C[m][n] + sum_k B[k][n]
// so the grand total of every 64-float chunk is preserved independent of the
// exact lane->B mapping. 2048 / 64 = 32 WMMA accumulations, then a 16-lane
// shuffle sum over D row 0 (C VGPR0, lanes 0..15).
// ---------------------------------------------------------------------------
__global__ __launch_bounds__(32)
void final_reduce_wmma(const float* __restrict__ part,
                       float*       __restrict__ out,
                       int n) {
    const int lane = threadIdx.x;           // 0..31, EXEC all ones (no divergence)
    v2f a; a.x = 1.0f; a.y = 1.0f;          // all-ones A matrix (16x4 f32)
    v8f c = {};                             // 16x16 f32 accumulator
    for (int base = 0; base < n; base += 64) {
        v2f b;
        b.x = part[base + lane];
        b.y = part[base + 32 + lane];
        c = __builtin_amdgcn_wmma_f32_16x16x4_f32(
                /*neg_a=*/false, a, /*neg_b=*/false, b,
                /*c_mod=*/(short)0, c, /*reuse_a=*/false, /*reuse_b=*/false);
    }
    // D[0][n] lives in c[0], lanes 0..15 (M=0, N=lane). Sum within 16-lane group.
    float v = c[0];
    v += __shfl_xor(v, 8, 32);
    v += __shfl_xor(v, 4, 32);
    v += __shfl_xor(v, 2, 32);
    v += __shfl_xor(v, 1, 32);
    if (lane == 0) out[0] = v;
}

extern "C" void kernel_launch(void* const* d_in, const int* in_sizes, int n_in,
                              void* d_out, int out_size, void* d_ws, size_t ws_size,
                              hipStream_t stream) {
    const float* logits = (const float*)d_in[0];   // [B, C] fp32
    const int*   labels = (const int*)d_in[1];     // [B] int
    float*       ws     = (float*)d_ws;            // per-row losses (kB floats)
    float*       res    = (float*)d_out;           // scalar fp32

    rowloss_kernel<<<kB, kBlock, 0, stream>>>(logits, labels, ws);
    final_reduce_wmma<<<1, 32, 0, stream>>>(ws, res, kB);   // kB % 64 == 0
}